// EchoLayer_28595892256913
// MI455X (gfx1250) — compile-verified
//
#include <hip/hip_runtime.h>
#include <math.h>

typedef float f4  __attribute__((ext_vector_type(4)));
typedef float v2f __attribute__((ext_vector_type(2)));
typedef float v8f __attribute__((ext_vector_type(8)));

#define B_      16
#define SEQ_    2048
#define D_      1024
#define HALF_   512
#define NPAT_   4096
#define PLEN_   32
#define SIM_    64
#define NPATCH_ 64                 /* SEQ_/PLEN_ */
#define ROWS_   (B_*NPATCH_)       /* 1024      */
#define KTOT_   (PLEN_*HALF_)      /* 16384     */

// ---------------------------------------------------------------------------
// Zero the first half of padding_out  ([B,2048] zeros, [B,2048] payload)
__global__ void k_zero_pad(float* __restrict__ pad) {
    int i = blockIdx.x * blockDim.x + threadIdx.x;          // 0 .. 32767
    pad[(i >> 11) * (2 * SEQ_) + (i & 2047)] = 0.0f;
}

// ---------------------------------------------------------------------------
// Streaming copy of src[:, :, :512] -> out_full[:, :, :512]  (non-temporal)
__global__ void k_copy_half(const f4* __restrict__ src4, f4* __restrict__ out4) {
    int i = blockIdx.x * blockDim.x + threadIdx.x;          // 0 .. 16*2048*128-1
    int row = i >> 7;                                       // global seq row
    int col = i & 127;                                      // float4 col in first half
    int idx = row * 256 + col;                              // row stride = 1024 floats
    f4 v = __builtin_nontemporal_load(src4 + idx);
    __builtin_nontemporal_store(v, out4 + idx);
}

// ---------------------------------------------------------------------------
// gpw_sum[h] = mean_j gp_w[h,j]; ws[512] = mean(gp_b)
__global__ void k_prep(const float* __restrict__ gp_w, const float* __restrict__ gp_b,
                       float* __restrict__ ws) {
    int h = threadIdx.x;
    if (h < HALF_) {
        float s = 0.f;
#pragma unroll
        for (int j = 0; j < 10; ++j) s += gp_w[h * 10 + j];
        ws[h] = s * 0.1f;
    }
    if (h == 0) {
        float s = 0.f;
#pragma unroll
        for (int j = 0; j < 10; ++j) s += gp_b[j];
        ws[HALF_] = s * 0.1f;
    }
}

// ---------------------------------------------------------------------------
// low0[row] = x[0,row,512:] . gpw_sum + gpb_mean   (batch 0 only; wave per row)
__global__ void k_low(const float* __restrict__ src, const float* __restrict__ ws,
                      float* __restrict__ low0) {
    int wave = (blockIdx.x * blockDim.x + threadIdx.x) >> 5;   // 0..2047
    int lane = threadIdx.x & 31;
    const float* xr = src + wave * D_ + HALF_;                  // b==0
    float acc = 0.f;
#pragma unroll
    for (int j = 0; j < 16; ++j) {
        int h = lane + j * 32;
        acc += xr[h] * ws[h];
    }
#pragma unroll
    for (int off = 16; off > 0; off >>= 1) acc += __shfl_down(acc, off, 32);
    if (lane == 0) low0[wave] = acc + ws[HALF_];
}

// ---------------------------------------------------------------------------
// Per patch (batch 0): scores over 4096 patterns, iterative top-64 (desc,
// smallest-index tie-break like jax.lax.top_k), gather selected[p][s][l].
__global__ void k_topk(const float* __restrict__ pool, const float* __restrict__ low0,
                       float* __restrict__ selected) {
    __shared__ float sc[NPAT_];
    __shared__ float rv[256];
    __shared__ int   ri[256];
    __shared__ float lw[PLEN_];
    __shared__ int   sel[SIM_];
    int p = blockIdx.x;
    int tid = threadIdx.x;
    if (tid < PLEN_) lw[tid] = low0[p * PLEN_ + tid];
    __syncthreads();
    for (int n = tid; n < NPAT_; n += 256) {
        const float* pr = pool + n * PLEN_;
        float a = 0.f;
#pragma unroll
        for (int l = 0; l < PLEN_; ++l) a += lw[l] * pr[l];
        sc[n] = a;
    }
    __syncthreads();
    for (int s = 0; s < SIM_; ++s) {
        float bv = -INFINITY; int bi = NPAT_;
        for (int n = tid; n < NPAT_; n += 256) {           // ascending scan keeps
            float v = sc[n];                               // smallest index on tie
            if (v > bv) { bv = v; bi = n; }
        }
        rv[tid] = bv; ri[tid] = bi;
        __syncthreads();
        for (int st = 128; st > 0; st >>= 1) {
            if (tid < st) {
                float v2 = rv[tid + st]; int i2 = ri[tid + st];
                if (v2 > rv[tid] || (v2 == rv[tid] && i2 < ri[tid])) {
                    rv[tid] = v2; ri[tid] = i2;
                }
            }
            __syncthreads();
        }
        if (tid == 0) { sel[s] = ri[0]; sc[ri[0]] = -INFINITY; }
        __syncthreads();
    }
    for (int e = tid; e < SIM_ * PLEN_; e += 256) {        // e = s*32 + l
        int s = e >> 5, l = e & 31;
        selected[p * (SIM_ * PLEN_) + e] = pool[sel[s] * PLEN_ + l];
    }
}

// ---------------------------------------------------------------------------
// logits[1024,64] = X[1024,16384] @ fuse_w + fuse_b   via V_WMMA_F32_16X16X4_F32
// Block: 128 thr (4 waves) -> 64 rows; each wave owns one 16x64 C strip.
// LDS layouts chosen so every WMMA fragment pair is one aligned ds_load_b64:
//   As[row][k]  row stride 34 (even, conflict-free), pair (k,k+1) contiguous
//   BsT[col][k] transposed,  stride 34, pair (k,k+1) contiguous
__global__ void k_logits(const float* __restrict__ src, const float* __restrict__ fuse_w,
                         const float* __restrict__ fuse_b, float* __restrict__ route) {
    __shared__ float As[64][34];
    __shared__ float BsT[64][34];
    int tid  = threadIdx.x;          // 0..127
    int w    = tid >> 5, lane = tid & 31;
    int m0   = blockIdx.x * 64;
    int l15  = lane & 15;
    int koff = (lane >> 4) << 1;     // lanes>=16 carry K+2/K+3 (f32 A layout)

    v8f acc[4];
#pragma unroll
    for (int n = 0; n < 4; ++n) acc[n] = (v8f){0,0,0,0,0,0,0,0};

    int arow  = tid >> 1;            // A loader: row 0..63
    int ahalf = tid & 1;             // which 16-K half
    int r  = m0 + arow;
    int bb = r >> 6, pp = r & 63;
    int bkk = tid >> 2;              // B loader: K-row 0..31
    int bq  = tid & 3;               // 16-col quarter

    for (int kc = 0; kc < KTOT_; kc += 32) {
        {   // A tile: 64 rows x 32 K   (k = l*512 + h, chunk stays inside one l)
            int k0 = kc + ahalf * 16;
            int l  = k0 >> 9, h = k0 & 511;
            const f4* g = (const f4*)(src + ((bb * SEQ_ + pp * PLEN_ + l) * D_ + HALF_ + h));
#pragma unroll
            for (int j = 0; j < 4; ++j) {
                f4 v = g[j];
#pragma unroll
                for (int c = 0; c < 4; ++c) As[arow][ahalf * 16 + j * 4 + c] = v[c];
            }
            if (kc + 32 < KTOT_) {   // prefetch next K-chunk (global_prefetch_b8)
                int k1 = kc + 32 + ahalf * 16;
                int ln = k1 >> 9, hn = k1 & 511;
                __builtin_prefetch(src + ((bb * SEQ_ + pp * PLEN_ + ln) * D_ + HALF_ + hn), 0, 0);
            }
        }
        {   // B tile: 32 K x 64 cols -> transposed into BsT[col][k]
            const f4* g = (const f4*)(fuse_w + (kc + bkk) * SIM_ + bq * 16);
#pragma unroll
            for (int j = 0; j < 4; ++j) {
                f4 v = g[j];
#pragma unroll
                for (int c = 0; c < 4; ++c) BsT[bq * 16 + j * 4 + c][bkk] = v[c];
            }
        }
        __syncthreads();
#pragma unroll
        for (int kk = 0; kk < 32; kk += 4) {
            int kb = kk + koff;                                   // even -> 8B aligned
            v2f Af = *(const v2f*)&As[w * 16 + l15][kb];
#pragma unroll
            for (int n = 0; n < 4; ++n) {
                v2f Bf = *(const v2f*)&BsT[n * 16 + l15][kb];
                acc[n] = __builtin_amdgcn_wmma_f32_16x16x4_f32(
                    false, Af, false, Bf, (short)0, acc[n], false, false);
            }
        }
        __syncthreads();
    }
    // C layout: VGPR j -> M=j (lanes 0-15) / M=j+8 (lanes 16-31), N = lane&15
#pragma unroll
    for (int n = 0; n < 4; ++n)
#pragma unroll
        for (int j = 0; j < 8; ++j) {
            int rowt = j + ((lane >> 4) << 3);
            int col  = n * 16 + l15;
            route[(m0 + w * 16 + rowt) * SIM_ + col] = acc[n][j] + fuse_b[col];
        }
}

// ---------------------------------------------------------------------------
// In-place softmax over 64 logits per row; one wave per row.
__global__ void k_softmax(float* __restrict__ route) {
    int row  = (blockIdx.x * blockDim.x + threadIdx.x) >> 5;   // 0..1023
    int lane = threadIdx.x & 31;
    float* rp = route + row * SIM_;
    float x0 = rp[lane], x1 = rp[lane + 32];
    float m = fmaxf(x0, x1);
#pragma unroll
    for (int off = 16; off > 0; off >>= 1) m = fmaxf(m, __shfl_xor(m, off, 32));
    float e0 = __expf(x0 - m), e1 = __expf(x1 - m);
    float s = e0 + e1;
#pragma unroll
    for (int off = 16; off > 0; off >>= 1) s += __shfl_xor(s, off, 32);
    float inv = 1.0f / s;
    rp[lane] = e0 * inv;
    rp[lane + 32] = e1 * inv;
}

// ---------------------------------------------------------------------------
// Per (b,p): Fuse[32,64] = route*selected (LDS), padding = row-sum(Fuse),
// out = Fuse @ recover_w + recover_b  via f32 WMMA. 8 waves: wave owns a
// 16x128 C strip (mt = w>>2, nq = w&3). recover_w staged TRANSPOSED in LDS
// (BsT2[col][k], stride 18 -> b64 fragment loads, bank-disjoint).
__global__ void k_out(const float* __restrict__ route, const float* __restrict__ selected,
                      const float* __restrict__ rw, const float* __restrict__ rb,
                      float* __restrict__ out, float* __restrict__ pad) {
    __shared__ float Fs[PLEN_][66];          // stride 66 (even) -> aligned pairs
    __shared__ float BsT2[HALF_][18];        // 16 K rows per chunk, transposed
    int bp  = blockIdx.x;
    int b   = bp >> 6, p = bp & 63;
    int tid = threadIdx.x;
    int w   = tid >> 5, lane = tid & 31;
    int l15 = lane & 15;
    int koff = (lane >> 4) << 1;
    int mt  = w >> 2, nq = w & 3;

    const float* rt = route + bp * SIM_;
    const float* sl = selected + p * (SIM_ * PLEN_);

    for (int e = tid; e < PLEN_ * SIM_; e += 256) {        // e = l*64 + s
        int l = e >> 6, s = e & 63;
        Fs[l][s] = rt[s] * sl[s * PLEN_ + l];
    }
    __syncthreads();
    if (tid < PLEN_) {                                     // padding row-sums
        float a = 0.f;
#pragma unroll
        for (int s = 0; s < SIM_; ++s) a += Fs[tid][s];
        pad[b * (2 * SEQ_) + SEQ_ + p * PLEN_ + tid] = a;
    }

    v8f acc[8];
#pragma unroll
    for (int nt = 0; nt < 8; ++nt) acc[nt] = (v8f){0,0,0,0,0,0,0,0};

    for (int kc = 0; kc < SIM_; kc += 16) {
        __syncthreads();
        for (int f = tid; f < 16 * 128; f += 256) {        // stage 16x512 of rw (transposed)
            int kk = f >> 7, c4 = f & 127;
            f4 v = ((const f4*)rw)[(kc + kk) * 128 + c4];
#pragma unroll
            for (int c = 0; c < 4; ++c) BsT2[c4 * 4 + c][kk] = v[c];
        }
        __syncthreads();
#pragma unroll
        for (int kk = 0; kk < 16; kk += 4) {
            int kb = kk + koff;                            // even -> 8B aligned
            v2f Af = *(const v2f*)&Fs[mt * 16 + l15][kc + kb];
#pragma unroll
            for (int nt = 0; nt < 8; ++nt) {
                int col = nq * 128 + nt * 16 + l15;
                v2f Bf = *(const v2f*)&BsT2[col][kb];
                acc[nt] = __builtin_amdgcn_wmma_f32_16x16x4_f32(
                    false, Af, false, Bf, (short)0, acc[nt], false, false);
            }
        }
    }
#pragma unroll
    for (int nt = 0; nt < 8; ++nt)
#pragma unroll
        for (int j = 0; j < 8; ++j) {
            int l = mt * 16 + j + ((lane >> 4) << 3);
            int h = nq * 128 + nt * 16 + l15;
            out[(b * SEQ_ + p * PLEN_ + l) * D_ + HALF_ + h] = acc[nt][j] + rb[h];
        }
}

// ---------------------------------------------------------------------------
extern "C" void kernel_launch(void* const* d_in, const int* in_sizes, int n_in,
                              void* d_out, int out_size, void* d_ws, size_t ws_size,
                              hipStream_t stream) {
    (void)in_sizes; (void)n_in; (void)out_size; (void)ws_size;
    const float* src    = (const float*)d_in[0];
    const float* pool   = (const float*)d_in[1];
    const float* fuse_w = (const float*)d_in[2];
    const float* fuse_b = (const float*)d_in[3];
    const float* rec_w  = (const float*)d_in[4];
    const float* rec_b  = (const float*)d_in[5];
    const float* gp_w   = (const float*)d_in[6];
    const float* gp_b   = (const float*)d_in[7];

    float* out_full = (float*)d_out;                       // [16,2048,1024]
    float* pad_out  = out_full + (size_t)B_ * SEQ_ * D_;   // [16,4096]

    // workspace layout (floats): gpw_sum[512]+mean | low0[2048] | selected[64*64*32] | route[1024*64]
    float* W        = (float*)d_ws;
    float* low0     = W + 1024;
    float* selected = W + 4096;
    float* route    = W + 4096 + NPATCH_ * SIM_ * PLEN_;

    k_zero_pad <<<128,   256, 0, stream>>>(pad_out);
    k_copy_half<<<16384, 256, 0, stream>>>((const f4*)src, (f4*)out_full);
    k_prep     <<<1,     512, 0, stream>>>(gp_w, gp_b, W);
    k_low      <<<256,   256, 0, stream>>>(src, W, low0);
    k_topk     <<<NPATCH_, 256, 0, stream>>>(pool, low0, selected);
    k_logits   <<<ROWS_ / 64, 128, 0, stream>>>(src, fuse_w, fuse_b, route);
    k_softmax  <<<ROWS_ / 8, 256, 0, stream>>>(route);
    k_out      <<<ROWS_, 256, 0, stream>>>(route, selected, rec_w, rec_b, out_full, pad_out);
}